// GINNet_9208409883137
// MI455X (gfx1250) — compile-verified
//
#include <hip/hip_runtime.h>
#include <hip/hip_bf16.h>
#include <math.h>

#define N_NODES  50000
#define N_EDGES  800000
#define DIM      64
#define N_GRAPHS 512
#define GIN_LAYERS 3
#define N_CLASSES 10

typedef __attribute__((ext_vector_type(2))) float v2f;
typedef __attribute__((ext_vector_type(8))) float v8f;

// ---------------------------------------------------------------------------
// Edge scatter-add: agg[dst] += h[src]   (agg pre-initialized to h => h + sum)
// 16 lanes per edge, float4 per lane. h is L2-resident (12.8MB << 192MB L2),
// so this is L2 atomic-throughput bound; use native global_atomic_add_f32.
// ---------------------------------------------------------------------------
__global__ __launch_bounds__(256) void gin_scatter_add(
    const float* __restrict__ h, const int* __restrict__ src,
    const int* __restrict__ dst, float* __restrict__ agg)
{
    long long idx = (long long)blockIdx.x * 256 + threadIdx.x;
    int e = (int)(idx >> 4);
    int c = (int)(idx & 15);
    if (e >= N_EDGES) return;
    int s = src[e];
    int d = dst[e];
    const float4 v = *(const float4*)(h + (size_t)s * DIM + c * 4);
    float* p = agg + (size_t)d * DIM + c * 4;
    unsafeAtomicAdd(p + 0, v.x);
    unsafeAtomicAdd(p + 1, v.y);
    unsafeAtomicAdd(p + 2, v.z);
    unsafeAtomicAdd(p + 3, v.w);
}

// ---------------------------------------------------------------------------
// out[M x 64] = act(A[M x 64] @ W[64 x 64] + bias)   via V_WMMA_F32_16X16X4_F32
// 8 waves / block, each wave owns a 16-row tile.
// W staged in LDS in *fragment-major* order: Wf[kstep][ntile][lane] is the
// per-lane v2f B-fragment {W[k][col], W[k+1][col]} (k = 4*kstep + 2*(lane>>4),
// col = 16*ntile + (lane&15)).  Each B fragment is then one ds_load_b64 into
// an even-aligned VGPR pair -- no repacking movs, bank-conflict-free.
// 16 K-steps x 4 N-tiles = 64 f32 WMMAs per wave, fp32 end-to-end.
// ---------------------------------------------------------------------------
__global__ __launch_bounds__(256) void gemm_wmma(
    const float* __restrict__ A, const float* __restrict__ W,
    const float* __restrict__ bias, float* __restrict__ out,
    int M, int relu)
{
    __shared__ v2f  Wf[16 * 4 * 32];   // 2048 fragments = 16KB
    __shared__ float bs[64];
    int t = threadIdx.x;
#pragma unroll
    for (int i = 0; i < 8; ++i) {
        int fi    = t + 256 * i;       // 0..2047
        int kstep = fi >> 7;
        int rem   = fi & 127;
        int ntile = rem >> 5;
        int l     = rem & 31;
        int col   = ntile * 16 + (l & 15);
        int k     = kstep * 4 + (l >> 4) * 2;
        v2f w;
        w.x = W[k * 64 + col];
        w.y = W[(k + 1) * 64 + col];
        Wf[fi] = w;
    }
    if (t < 64) bs[t] = bias[t];
    __syncthreads();

    int lane  = t & 31;
    int tile  = blockIdx.x * 8 + (t >> 5);
    if (tile * 16 >= M) return;                // wave-uniform: EXEC stays all-1s
    int m0    = tile << 4;
    int colb  = lane & 15;
    int khalf = lane >> 4;                     // 0: K=kb..kb+1, 1: K=kb+2..kb+3
    const float* Arow = A + (size_t)(m0 + colb) * DIM;
    const v2f* wf = Wf + lane;

    v8f c0 = {}, c1 = {}, c2 = {}, c3 = {};
#pragma unroll
    for (int ks = 0; ks < 16; ++ks) {
        float2 av = *(const float2*)(Arow + ks * 4 + khalf * 2);
        v2f a; a.x = av.x; a.y = av.y;

        v2f b0 = wf[ks * 128 +  0];
        v2f b1 = wf[ks * 128 + 32];
        v2f b2 = wf[ks * 128 + 64];
        v2f b3 = wf[ks * 128 + 96];

        c0 = __builtin_amdgcn_wmma_f32_16x16x4_f32(false, a, false, b0, (short)0, c0, false, false);
        c1 = __builtin_amdgcn_wmma_f32_16x16x4_f32(false, a, false, b1, (short)0, c1, false, false);
        c2 = __builtin_amdgcn_wmma_f32_16x16x4_f32(false, a, false, b2, (short)0, c2, false, false);
        c3 = __builtin_amdgcn_wmma_f32_16x16x4_f32(false, a, false, b3, (short)0, c3, false, false);
    }

#pragma unroll
    for (int r = 0; r < 8; ++r) {
        int orow = m0 + r + khalf * 8;         // C/D layout: VGPR r -> M=r / M=r+8
        float* po = out + (size_t)orow * DIM + colb;
        float v0 = c0[r] + bs[colb +  0];
        float v1 = c1[r] + bs[colb + 16];
        float v2 = c2[r] + bs[colb + 32];
        float v3 = c3[r] + bs[colb + 48];
        if (relu) {
            v0 = fmaxf(v0, 0.f); v1 = fmaxf(v1, 0.f);
            v2 = fmaxf(v2, 0.f); v3 = fmaxf(v3, 0.f);
        }
        po[0] = v0; po[16] = v1; po[32] = v2; po[48] = v3;
    }
}

// ---------------------------------------------------------------------------
// Per-graph max pool via order-preserving uint encoding + atomicMax
// ---------------------------------------------------------------------------
__device__ __forceinline__ unsigned encodeF(float f) {
    unsigned u = __float_as_uint(f);
    return (u & 0x80000000u) ? ~u : (u | 0x80000000u);
}
__device__ __forceinline__ float decodeF(unsigned u) {
    unsigned v = (u & 0x80000000u) ? (u & 0x7FFFFFFFu) : ~u;
    return __uint_as_float(v);
}

__global__ __launch_bounds__(256) void pool_init(unsigned* genc)
{
    int i = blockIdx.x * 256 + threadIdx.x;
    if (i < N_GRAPHS * DIM) genc[i] = 0x007FFFFFu;   // encodeF(-inf)
}

__global__ __launch_bounds__(256) void pool_max(
    const float* __restrict__ h, const int* __restrict__ batch,
    unsigned* __restrict__ genc)
{
    int idx = blockIdx.x * 256 + threadIdx.x;
    if (idx >= N_NODES * DIM) return;
    int n = idx >> 6;
    int d = idx & 63;
    atomicMax(&genc[(size_t)batch[n] * DIM + d], encodeF(h[idx]));
}

__global__ __launch_bounds__(256) void pool_decode(
    const unsigned* __restrict__ genc, float* __restrict__ g)
{
    int i = blockIdx.x * 256 + threadIdx.x;
    if (i < N_GRAPHS * DIM) g[i] = decodeF(genc[i]);
}

// ---------------------------------------------------------------------------
// FC2 (64 -> 10) + log_softmax; one wave32 per graph, shuffle reductions
// ---------------------------------------------------------------------------
__global__ __launch_bounds__(32) void fc_head(
    const float* __restrict__ g1, const float* __restrict__ W,
    const float* __restrict__ b, float* __restrict__ out)
{
    int row  = blockIdx.x;
    int lane = threadIdx.x;
    float logit = -INFINITY;
    if (lane < N_CLASSES) {
        float acc = b[lane];
        const float* gr = g1 + (size_t)row * DIM;
#pragma unroll
        for (int k = 0; k < DIM; ++k)
            acc = fmaf(gr[k], W[k * N_CLASSES + lane], acc);
        logit = acc;
    }
    float m = logit;
    for (int off = 16; off > 0; off >>= 1)
        m = fmaxf(m, __shfl_xor(m, off, 32));
    float e = (lane < N_CLASSES) ? expf(logit - m) : 0.0f;
    float s = e;
    for (int off = 16; off > 0; off >>= 1)
        s += __shfl_xor(s, off, 32);
    if (lane < N_CLASSES)
        out[(size_t)row * N_CLASSES + lane] = logit - m - logf(s);
}

// ---------------------------------------------------------------------------
extern "C" void kernel_launch(void* const* d_in, const int* in_sizes, int n_in,
                              void* d_out, int out_size, void* d_ws, size_t ws_size,
                              hipStream_t stream) {
    (void)in_sizes; (void)n_in; (void)out_size; (void)ws_size;

    const float* x      = (const float*)d_in[0];
    const float* convW1 = (const float*)d_in[1];
    const float* convb1 = (const float*)d_in[2];
    const float* convW2 = (const float*)d_in[3];
    const float* convb2 = (const float*)d_in[4];
    const float* fcW1   = (const float*)d_in[5];
    const float* fcb1   = (const float*)d_in[6];
    const float* fcW2   = (const float*)d_in[7];
    const float* fcb2   = (const float*)d_in[8];
    const int*   edge   = (const int*)d_in[9];
    const int*   batch  = (const int*)d_in[10];
    const int* src = edge;
    const int* dst = edge + N_EDGES;

    char* ws = (char*)d_ws;
    const size_t szH = (size_t)N_NODES * DIM * sizeof(float);   // 12.8 MB
    const size_t szG = (size_t)N_GRAPHS * DIM * sizeof(float);  // 128 KB
    float*    hA   = (float*)(ws);
    float*    hB   = (float*)(ws + szH);
    unsigned* genc = (unsigned*)(ws + 2 * szH);
    float*    g    = (float*)(ws + 2 * szH + szG);
    float*    g1   = (float*)(ws + 2 * szH + 2 * szG);

    hipMemcpyAsync(hA, x, szH, hipMemcpyDeviceToDevice, stream);

    const int nodeTiles  = N_NODES / 16;                // 3125
    const int gemmBlocks = (nodeTiles + 7) / 8;         // 391
    const int scatGrid   = (int)(((long long)N_EDGES * 16 + 255) / 256);

    float* cur = hA;
    float* oth = hB;
    for (int l = 0; l < GIN_LAYERS; ++l) {
        hipMemcpyAsync(oth, cur, szH, hipMemcpyDeviceToDevice, stream);
        gin_scatter_add<<<scatGrid, 256, 0, stream>>>(cur, src, dst, oth);
        gemm_wmma<<<gemmBlocks, 256, 0, stream>>>(
            oth, convW1 + (size_t)l * DIM * DIM, convb1 + (size_t)l * DIM,
            cur, N_NODES, 1);
        gemm_wmma<<<gemmBlocks, 256, 0, stream>>>(
            cur, convW2 + (size_t)l * DIM * DIM, convb2 + (size_t)l * DIM,
            oth, N_NODES, 0);
        float* tmp = cur; cur = oth; oth = tmp;
    }

    pool_init<<<(N_GRAPHS * DIM + 255) / 256, 256, 0, stream>>>(genc);
    pool_max<<<(N_NODES * DIM + 255) / 256, 256, 0, stream>>>(cur, batch, genc);
    pool_decode<<<(N_GRAPHS * DIM + 255) / 256, 256, 0, stream>>>(genc, g);

    gemm_wmma<<<(N_GRAPHS / 16 + 7) / 8, 256, 0, stream>>>(
        g, fcW1, fcb1, g1, N_GRAPHS, 1);
    fc_head<<<N_GRAPHS, 32, 0, stream>>>(g1, fcW2, fcb2, (float*)d_out);
}